// MRFLoss_27135603376138
// MI455X (gfx1250) — compile-verified
//
#include <hip/hip_runtime.h>
#include <math.h>

// MRF loss: softmax(5) -> segmented {sum p, sum p^2} over 4096 superpixels + entropy.
// Memory-bound: ~112MB traffic -> ~4.8us floor @ 23.3TB/s.
// cnt[seg] derived from sum_c s1[seg][c] (softmax rows sum to 1), so the
// per-block LDS histogram is exactly 4096*10 floats = 163840B (fits CDNA5 LDS).
// Points processed in pairs: 2 rows = 48B = 3x global_load_b128, 16B aligned.

#define NUM_SP    4096
#define HIST_F    (NUM_SP * 10)     // per segment: s1[0..4], s2[0..4]
#define WS_ENT    HIST_F            // entropy accumulator slot
#define WS_FLOATS (HIST_F + 1)

typedef float v2f __attribute__((ext_vector_type(2)));
typedef float v8f __attribute__((ext_vector_type(8)));

// Wave32 sum via V_WMMA_F32_16X16X4_F32 (all-ones B is layout-agnostic):
// D[m][n] = v(lane m) + v(lane m+16); lanes 0-15 see rows 0-7 in d[0..7],
// lanes 16-31 see rows 8-15; sum the 8 regs, fold across half-waves.
__device__ __forceinline__ float wave32_sum_wmma(float v) {
  v2f a; a[0] = v;    a[1] = 0.0f;
  v2f b; b[0] = 1.0f; b[1] = 1.0f;
  v8f c = {};
  v8f d = __builtin_amdgcn_wmma_f32_16x16x4_f32(false, a, false, b,
                                                (short)0, c, false, false);
  float t = d[0] + d[1] + d[2] + d[3] + d[4] + d[5] + d[6] + d[7];
  t += __shfl_xor(t, 16, 32);
  return t;
}

// softmax(5) -> entropy accumulate + 10x ds_add_f32 into the LDS histogram
__device__ __forceinline__ void process_point(float x0, float x1, float x2,
                                              float x3, float x4, int sg,
                                              float* hist, float& ent) {
  float m  = fmaxf(fmaxf(fmaxf(x0, x1), fmaxf(x2, x3)), x4);
  float e0 = __expf(x0 - m), e1 = __expf(x1 - m), e2 = __expf(x2 - m),
        e3 = __expf(x3 - m), e4 = __expf(x4 - m);
  float inv = __builtin_amdgcn_rcpf(e0 + e1 + e2 + e3 + e4);  // v_rcp_f32
  float p0 = e0 * inv, p1 = e1 * inv, p2 = e2 * inv,
        p3 = e3 * inv, p4 = e4 * inv;
  ent -= p0 * __logf(p0 + 1e-8f) + p1 * __logf(p1 + 1e-8f)
       + p2 * __logf(p2 + 1e-8f) + p3 * __logf(p3 + 1e-8f)
       + p4 * __logf(p4 + 1e-8f);
  float* h = hist + sg * 10;
  atomicAdd(h + 0, p0);        atomicAdd(h + 1, p1);
  atomicAdd(h + 2, p2);        atomicAdd(h + 3, p3);
  atomicAdd(h + 4, p4);
  atomicAdd(h + 5, p0 * p0);   atomicAdd(h + 6, p1 * p1);
  atomicAdd(h + 7, p2 * p2);   atomicAdd(h + 8, p3 * p3);
  atomicAdd(h + 9, p4 * p4);
}

__global__ void mrf_zero(float* __restrict__ ws) {
  int i = blockIdx.x * 256 + threadIdx.x;
  if (i < WS_FLOATS) ws[i] = 0.0f;
}

__global__ __launch_bounds__(256) void mrf_main(const float* __restrict__ logits,
                                                const int* __restrict__ seg,
                                                float* __restrict__ ws, int n) {
  __shared__ __align__(16) float hist[HIST_F];   // 163840 bytes
  {
    const float4 z = make_float4(0.f, 0.f, 0.f, 0.f);
    for (int i = threadIdx.x * 4; i < HIST_F; i += 1024)
      *(float4*)&hist[i] = z;                    // ds_store_b128
  }
  __syncthreads();

  float ent = 0.0f;
  const int gid    = blockIdx.x * 256 + threadIdx.x;
  const int stride = gridDim.x * 256;
  const int npair  = n >> 1;

  for (int j = gid; j < npair; j += stride) {
    const float4* rp = (const float4*)(logits + (size_t)j * 12u);  // 48B, 16B aligned
    float4 r0 = rp[0], r1 = rp[1], r2 = rp[2];                     // 3x global_load_b128
    int2 sg = *(const int2*)(seg + (size_t)j * 2u);                // global_load_b64
    process_point(r0.x, r0.y, r0.z, r0.w, r1.x, sg.x, hist, ent);  // row 2j   (class 5 = r1.y unused)
    process_point(r1.z, r1.w, r2.x, r2.y, r2.z, sg.y, hist, ent);  // row 2j+1 (class 5 = r2.w unused)
  }
  if ((n & 1) && gid == 0) {                                       // odd-N tail
    const float* row = logits + (size_t)(n - 1) * 6u;
    process_point(row[0], row[1], row[2], row[3], row[4], seg[n - 1], hist, ent);
  }

  // wave32 entropy reduction; one global float atomic per wave
  for (int off = 16; off > 0; off >>= 1) ent += __shfl_xor(ent, off, 32);
  if ((threadIdx.x & 31) == 0) unsafeAtomicAdd(&ws[WS_ENT], ent);

  __syncthreads();
  for (int i = threadIdx.x * 4; i < HIST_F; i += 1024) {
    float4 v = *(const float4*)&hist[i];                           // ds_load_b128
    unsafeAtomicAdd(&ws[i + 0], v.x);
    unsafeAtomicAdd(&ws[i + 1], v.y);
    unsafeAtomicAdd(&ws[i + 2], v.z);
    unsafeAtomicAdd(&ws[i + 3], v.w);
  }
}

__global__ __launch_bounds__(128) void mrf_final(const float* __restrict__ ws,
                                                 float* __restrict__ out, int n) {
  float smooth = 0.0f, nsp = 0.0f;
  for (int s = threadIdx.x; s < NUM_SP; s += 128) {
    const float* h = ws + s * 10;
    float a0 = h[0], a1 = h[1], a2 = h[2], a3 = h[3], a4 = h[4];
    float b0 = h[5], b1 = h[6], b2 = h[7], b3 = h[8], b4 = h[9];

    float cnt = rintf(a0 + a1 + a2 + a3 + a4);   // sum_c p_c == 1 per point
    if (cnt > 0.5f) nsp += 1.0f;
    float cc  = fmaxf(cnt, 1.0f);
    float cm1 = fmaxf(cnt - 1.0f, 1.0f);

    float mean, acc = 0.0f;
    mean = a0 / cc; acc += b0 - cnt * mean * mean;
    mean = a1 / cc; acc += b1 - cnt * mean * mean;
    mean = a2 / cc; acc += b2 - cnt * mean * mean;
    mean = a3 / cc; acc += b3 - cnt * mean * mean;
    mean = a4 / cc; acc += b4 - cnt * mean * mean;
    if (cnt >= 1.5f) smooth += acc / (5.0f * cm1);  // mean-over-classes variance
  }

  float S = wave32_sum_wmma(smooth);   // v_wmma_f32_16x16x4_f32
  float P = wave32_sum_wmma(nsp);

  __shared__ float red[8];
  int wave = threadIdx.x >> 5;
  if ((threadIdx.x & 31) == 0) { red[wave] = S; red[4 + wave] = P; }
  __syncthreads();
  if (threadIdx.x == 0) {
    float Sa = red[0] + red[1] + red[2] + red[3];
    float Pa = fmaxf(red[4] + red[5] + red[6] + red[7], 1.0f);
    float entropy = ws[WS_ENT] / (float)n;
    out[0] = 0.8f * (Sa / Pa) + 0.2f * entropy;
  }
}

extern "C" void kernel_launch(void* const* d_in, const int* in_sizes, int n_in,
                              void* d_out, int out_size, void* d_ws, size_t ws_size,
                              hipStream_t stream) {
  (void)n_in; (void)out_size; (void)ws_size;
  const float* logits = (const float*)d_in[0];   // (N, 6) f32
  const int*   seg    = (const int*)d_in[1];     // (N,) int
  float* out = (float*)d_out;
  float* ws  = (float*)d_ws;
  const int n = in_sizes[1];

  mrf_zero<<<(WS_FLOATS + 255) / 256, 256, 0, stream>>>(ws);
  mrf_main<<<256, 256, 0, stream>>>(logits, seg, ws, n);
  mrf_final<<<1, 128, 0, stream>>>(ws, out, n);
}